// SmoothAP_68745246540427
// MI455X (gfx1250) — compile-verified
//
#include <hip/hip_runtime.h>

typedef __attribute__((ext_vector_type(2))) float v2f;
typedef __attribute__((ext_vector_type(8))) float v8f;

#define NN 384
#define DD 128
#define ANNEAL_INV 100.0f

// ---------------------------------------------------------------------------
// Kernel 1: sim = preds @ preds^T via V_WMMA_F32_16X16X4_F32.
// One wave (32 lanes) per 16x16 output tile; K accumulated in steps of 4.
// A (16x4 f32): lanes 0-15 hold K=0,1 ; lanes 16-31 hold K=2,3 ; M = lane&15.
// B (4x16 f32): same split over K, N = lane&15 (B tile = preds rows at col base).
// D (16x16 f32): VGPR v -> M = v + 8*(lane>=16), N = lane&15.
// ---------------------------------------------------------------------------
__global__ __launch_bounds__(32) void sim_wmma_kernel(const float* __restrict__ preds,
                                                      float* __restrict__ sim) {
    const int bm   = blockIdx.x;       // row tile
    const int bn   = blockIdx.y;       // col tile
    const int lane = threadIdx.x;
    const int hi   = lane >> 4;        // 0: K pair {0,1}, 1: K pair {2,3}
    const int lr   = lane & 15;

    const float* arow = preds + (bm * 16 + lr) * DD + 2 * hi;
    const float* brow = preds + (bn * 16 + lr) * DD + 2 * hi;

    v8f acc = {};
    #pragma unroll
    for (int k0 = 0; k0 < DD; k0 += 4) {
        v2f a = *(const v2f*)(arow + k0);
        v2f b = *(const v2f*)(brow + k0);
        acc = __builtin_amdgcn_wmma_f32_16x16x4_f32(
            /*neg_a=*/false, a, /*neg_b=*/false, b,
            /*c_mod=*/(short)0, acc, /*reuse_a=*/false, /*reuse_b=*/false);
    }

    #pragma unroll
    for (int v = 0; v < 8; ++v) {
        sim[(bm * 16 + v + 8 * hi) * NN + bn * 16 + lr] = acc[v];
    }
}

// ---------------------------------------------------------------------------
// Kernel 2: for each row a, compute
//   rowpart[a] = (n_pos>1) ? ( sum_b (1 + Ipos_ab + pos_sum_ab*Ipos_ab)
//                                    / (1 + sum_all_ab) ) / n_pos : 0
// sim row and labels staged in LDS; inner c-loop is the 56M-sigmoid hot loop.
// ---------------------------------------------------------------------------
__global__ __launch_bounds__(256) void smoothap_row_kernel(const float* __restrict__ sim,
                                                           const int* __restrict__ labels,
                                                           float* __restrict__ rowpart) {
    const int a = blockIdx.x;
    const int t = threadIdx.x;

    __shared__ float s_sim[NN];
    __shared__ int   s_lab[NN];
    __shared__ float s_red[256];

    for (int c = t; c < NN; c += 256) {
        s_sim[c] = sim[a * NN + c];
        s_lab[c] = labels[c];
    }
    __syncthreads();

    const int lab_a = s_lab[a];

    // partial n_pos = sum_c Ipos[a,c]
    float npos_local = 0.0f;
    for (int c = t; c < NN; c += 256)
        npos_local += (s_lab[c] == lab_a && c != a) ? 1.0f : 0.0f;

    // partial sum over b of per-(a,b) AP term
    float term_local = 0.0f;
    for (int b = t; b < NN; b += 256) {
        const float sab = s_sim[b];
        float sum_all = 0.0f;
        float pos_sum = 0.0f;
        for (int c = 0; c < NN; ++c) {
            float x = (s_sim[c] - sab) * ANNEAL_INV;
            x = fminf(50.0f, fmaxf(-50.0f, x));
            float sg = 1.0f / (1.0f + __expf(-x));
            sg = (c == b) ? 0.0f : sg;                              // (1 - eye) over (b,c)
            sum_all += sg;
            pos_sum += (s_lab[c] == lab_a && c != a) ? sg : 0.0f;   // * Ipos[a,c]
        }
        const float ipos_ab = (s_lab[b] == lab_a && b != a) ? 1.0f : 0.0f;
        term_local += (1.0f + ipos_ab + pos_sum * ipos_ab) / (1.0f + sum_all);
    }

    // block reduction: terms
    s_red[t] = term_local;
    __syncthreads();
    for (int s = 128; s > 0; s >>= 1) {
        if (t < s) s_red[t] += s_red[t + s];
        __syncthreads();
    }
    const float term_total = s_red[0];
    __syncthreads();

    // block reduction: n_pos
    s_red[t] = npos_local;
    __syncthreads();
    for (int s = 128; s > 0; s >>= 1) {
        if (t < s) s_red[t] += s_red[t + s];
        __syncthreads();
    }

    if (t == 0) {
        const float n_pos = s_red[0] + 1.0f;
        rowpart[a] = (n_pos > 1.0f) ? (term_total / n_pos) : 0.0f;
    }
}

// ---------------------------------------------------------------------------
// Kernel 3: out[0] = 1 - (sum_a rowpart[a]) / N
// ---------------------------------------------------------------------------
__global__ __launch_bounds__(256) void finalize_kernel(const float* __restrict__ rowpart,
                                                       float* __restrict__ out) {
    __shared__ float s_red[256];
    const int t = threadIdx.x;
    float v = 0.0f;
    for (int i = t; i < NN; i += 256) v += rowpart[i];
    s_red[t] = v;
    __syncthreads();
    for (int s = 128; s > 0; s >>= 1) {
        if (t < s) s_red[t] += s_red[t + s];
        __syncthreads();
    }
    if (t == 0) out[0] = 1.0f - s_red[0] / (float)NN;
}

extern "C" void kernel_launch(void* const* d_in, const int* in_sizes, int n_in,
                              void* d_out, int out_size, void* d_ws, size_t ws_size,
                              hipStream_t stream) {
    (void)in_sizes; (void)n_in; (void)out_size; (void)ws_size;

    const float* preds  = (const float*)d_in[0];   // [384,128] f32, L2-normalized
    const int*   labels = (const int*)d_in[1];     // [384] int
    float*       out    = (float*)d_out;           // [1] f32

    float* sim     = (float*)d_ws;                 // 384*384 f32 = 576 KB
    float* rowpart = sim + (size_t)NN * NN;        // 384 f32

    dim3 gGemm(NN / 16, NN / 16);                  // 24 x 24 tiles
    sim_wmma_kernel<<<gGemm, 32, 0, stream>>>(preds, sim);
    smoothap_row_kernel<<<NN, 256, 0, stream>>>(sim, labels, rowpart);
    finalize_kernel<<<1, 256, 0, stream>>>(rowpart, out);
}